// CRFEntityExtractor_45260365365253
// MI455X (gfx1250) — compile-verified
//
#include <hip/hip_runtime.h>
#include <hip/hip_bf16.h>

typedef float v2f __attribute__((ext_vector_type(2)));
typedef float v8f __attribute__((ext_vector_type(8)));

namespace {
constexpr int Bn = 256, Tn = 512, Hn = 256, Kn = 32;
constexpr int CHUNK  = 64;             // timesteps staged per pipeline stage
constexpr int NCHUNK = Tn / CHUNK;     // 8
constexpr int EPITCH = 260;            // 256 + 4 pad -> bank-conflict-free strided reads
constexpr int WPITCH = 260;
constexpr int LPITCH = 33;             // 32 + 1 pad
constexpr int EMIS_F  = CHUNK * EPITCH;        // 16640 floats
constexpr int WT_F    = Kn * WPITCH;           // 8320 floats
constexpr int LOGIT_F = 2 * CHUNK * LPITCH;    // 4224 floats (double buffered)
constexpr size_t SMEM_BYTES =
    (size_t)(EMIS_F + WT_F + LOGIT_F) * 4 + 2 * CHUNK * 4 /*tags*/ + 2 * CHUNK * 4 /*mask*/;
}

// ---- cooperative coalesced load of one 64xH emissions chunk (+tags/mask) into LDS ----
__device__ __forceinline__ void load_chunk(const float* __restrict__ eb,
                                           const int* __restrict__ tg,
                                           const unsigned char* __restrict__ mk,
                                           float* emis, int* tagl, int* mskl,
                                           int c, int tid0, int nthreads) {
  const int base = c * CHUNK;
  const float4* __restrict__ src = (const float4*)(eb + (size_t)base * Hn);
  constexpr int V4_PER_ROW = Hn / 4;            // 64
  constexpr int TOTAL4 = CHUNK * V4_PER_ROW;    // 4096
  for (int i = tid0; i < TOTAL4; i += nthreads) {
    const int row = i >> 6, c4 = i & 63;
    *(float4*)(emis + row * EPITCH + c4 * 4) = src[i];
  }
  for (int i = tid0; i < CHUNK; i += nthreads) {
    tagl[i] = tg[base + i];
    mskl[i] = (int)mk[base + i];
  }
}

// ---- one 16x32 logit tile per wave via V_WMMA_F32_16X16X4_F32, H-loop depth 4 ----
__device__ __forceinline__ void gemm_chunk(const float* emis, const float* Wt,
                                           float b0v, float b1v,
                                           float* logit_buf, int wave, int lane) {
  const int half = lane >> 4;     // K-half of the A/B fragment
  const int lrow = lane & 15;     // M (A) / N (B) index
  const float* arow = emis + (wave * 16 + lrow) * EPITCH + 2 * half;
  const float* w0p  = Wt  + lrow        * WPITCH + 2 * half;   // N = 0..15
  const float* w1p  = Wt  + (16 + lrow) * WPITCH + 2 * half;   // N = 16..31
  v8f acc0 = {}, acc1 = {};
#pragma unroll
  for (int kk = 0; kk < Hn; kk += 4) {
    v2f a  = *(const v2f*)(arow + kk);
    v2f w0 = *(const v2f*)(w0p + kk);
    v2f w1 = *(const v2f*)(w1p + kk);
    acc0 = __builtin_amdgcn_wmma_f32_16x16x4_f32(false, a, false, w0, (short)0, acc0, false, false);
    acc1 = __builtin_amdgcn_wmma_f32_16x16x4_f32(false, a, false, w1, (short)0, acc1, false, false);
  }
  // C/D layout: VGPR r -> lanes 0-15: M=r ; lanes 16-31: M=r+8 ; N = lane%16 (+n_base)
#pragma unroll
  for (int r = 0; r < 8; ++r) {
    const int row = wave * 16 + r + 8 * half;
    logit_buf[row * LPITCH + lrow]      = acc0[r] + b0v;
    logit_buf[row * LPITCH + 16 + lrow] = acc1[r] + b1v;
  }
}

__global__ void __launch_bounds__(128)
crf_fused_kernel(const float* __restrict__ emissions, const float* __restrict__ W,
                 const float* __restrict__ bias, const float* __restrict__ start_t,
                 const float* __restrict__ trans, const float* __restrict__ end_t,
                 const int* __restrict__ tags, const unsigned char* __restrict__ mask,
                 float* __restrict__ out) {
  extern __shared__ float smem[];
  float* emis  = smem;                     // [CHUNK][EPITCH]
  float* Wt    = emis + EMIS_F;            // [Kn][WPITCH]  (W transposed)
  float* logit = Wt + WT_F;                // [2][CHUNK][LPITCH]
  int*   tagl  = (int*)(logit + LOGIT_F);  // [2][CHUNK]
  int*   mskl  = tagl + 2 * CHUNK;         // [2][CHUNK]

  const int b    = blockIdx.x;
  const int tid  = threadIdx.x;
  const int wave = tid >> 5;
  const int lane = tid & 31;

  const float* __restrict__ eb = emissions + (size_t)b * Tn * Hn;
  const int* __restrict__ tg   = tags + (size_t)b * Tn;
  const unsigned char* __restrict__ mk = mask + (size_t)b * Tn;

  // Stage W transposed (bank-friendly for B fragments): Wt[k][h] = W[h][k]
  for (int i = tid; i < Hn * Kn; i += 128) {
    const int h = i >> 5, k = i & 31;
    Wt[k * WPITCH + h] = W[i];
  }

  // Per-lane scan state (wave 0, lane = CRF state k)
  float tc[Kn];                 // transitions[j][lane], column resident in regs
  float alpha = 0.f, score = 0.f;
  int   prev  = 0;
  const float st_l = start_t[lane];
  const float et_l = end_t[lane];
  if (wave == 0) {
#pragma unroll
    for (int j = 0; j < Kn; ++j) tc[j] = trans[j * Kn + lane];
  }
  const float b0v = bias[lane & 15];
  const float b1v = bias[16 + (lane & 15)];

  // ---- prologue: stage chunk 0, compute its logits ----
  load_chunk(eb, tg, mk, emis, tagl, mskl, 0, tid, 128);
  __syncthreads();
  gemm_chunk(emis, Wt, b0v, b1v, logit, wave, lane);
  __syncthreads();

  for (int c = 0; c < NCHUNK; ++c) {
    const int buf = c & 1;
    if (wave == 0) {
      // -------- sequential CRF scan over this chunk (lane k = state k) --------
      float* lb = logit + buf * CHUNK * LPITCH;
      int*   tb = tagl + buf * CHUNK;
      int*   mb = mskl + buf * CHUNK;
      for (int tl = 0; tl < CHUNK; ++tl) {
        const float lg_lane = lb[tl * LPITCH + lane];
        const int   tagt    = tb[tl];
        const int   mt      = mb[tl];
        const float lg_tag  = lb[tl * LPITCH + tagt];   // uniform addr -> broadcast
        if (c == 0 && tl == 0) {
          alpha = st_l + lg_lane;
          prev  = tagt;
          score = start_t[tagt] + lg_tag;
        } else {
          float v[Kn];
          float m = -__builtin_inff();
#pragma unroll
          for (int j = 0; j < Kn; ++j) {
            v[j] = __shfl(alpha, j, 32) + tc[j];
            m = fmaxf(m, v[j]);
          }
          float s = 0.f;
#pragma unroll
          for (int j = 0; j < Kn; ++j) s += __expf(v[j] - m);
          const float nxt  = m + __logf(s) + lg_lane;
          const float step = trans[prev * Kn + tagt] + lg_tag;
          if (mt) {
            alpha  = nxt;
            score += step;
            prev   = tagt;
          }
        }
      }
    } else if (c + 1 < NCHUNK) {
      // -------- waves 1-3 prefetch next chunk while wave 0 scans --------
      load_chunk(eb, tg, mk, emis, tagl + (buf ^ 1) * CHUNK, mskl + (buf ^ 1) * CHUNK,
                 c + 1, tid - 32, 96);
    }
    __syncthreads();
    if (c + 1 < NCHUNK) {
      gemm_chunk(emis, Wt, b0v, b1v, logit + (buf ^ 1) * CHUNK * LPITCH, wave, lane);
    }
    __syncthreads();
  }

  if (wave == 0) {
    // logZ = logsumexp_k(alpha_k + end_k) via wave32 butterfly reduction
    const float vend = alpha + et_l;
    float m = vend;
#pragma unroll
    for (int off = 16; off; off >>= 1) m = fmaxf(m, __shfl_xor(m, off, 32));
    float s = __expf(vend - m);
#pragma unroll
    for (int off = 16; off; off >>= 1) s += __shfl_xor(s, off, 32);
    const float logz = m + __logf(s);
    score += end_t[prev];
    if (lane == 0) atomicAdd(out, score - logz);
  }
}

extern "C" void kernel_launch(void* const* d_in, const int* in_sizes, int n_in,
                              void* d_out, int out_size, void* d_ws, size_t ws_size,
                              hipStream_t stream) {
  const float* emissions = (const float*)d_in[0];
  const float* W         = (const float*)d_in[1];
  const float* bias      = (const float*)d_in[2];
  const float* start_t   = (const float*)d_in[3];
  const float* trans     = (const float*)d_in[4];
  const float* end_t     = (const float*)d_in[5];
  const int*   tags      = (const int*)d_in[6];
  const unsigned char* mask = (const unsigned char*)d_in[7];
  float* out = (float*)d_out;

  hipMemsetAsync(out, 0, sizeof(float), stream);
  crf_fused_kernel<<<Bn, 128, SMEM_BYTES, stream>>>(
      emissions, W, bias, start_t, trans, end_t, tags, mask, out);
}